// KMeansTrainableQuantizer_6296422056169
// MI455X (gfx1250) — compile-verified
//
#include <hip/hip_runtime.h>

// y = b + sum_i alpha_i * step(x - th_i), 15 thresholds, elementwise over 67.1M fp32.
// HBM-streaming kernel (roofline ~22us @ 23.3 TB/s). B128 non-temporal loads/stores,
// grid-stride x4 unroll, global_prefetch_b8.
//
// Per-threshold inner cost reduced to 2 VALU via the hardware clamp modifier:
//   step(x - t) = clamp(fma(x, BIG, c)),  BIG = 2^(23 - exponent(t)),
//   c = fma(-t, BIG, 1.0f)  (exact: t*BIG is an integer in [2^23, 2^24)).
// At x == t the fma gives exactly 1.0 (step=1, matching u>=0 semantics); at
// x = t - 1ulp it gives exactly 0.0; NaN clamps to 0 (matching NaN>=0 == false).

typedef float f4 __attribute__((ext_vector_type(4)));

#define NTHR 15
#define BLOCK 256
#define MAX_BLOCKS 4096

__device__ __forceinline__ float stepfn(float v, float big, float c) {
    // v_fma_f32 with clamp modifier (LLVM folds the min/max pair into clamp).
    return fminf(fmaxf(fmaf(v, big, c), 0.0f), 1.0f);
}

__device__ __forceinline__ float qmap1(float v,
                                       const float* __restrict__ big,
                                       const float* __restrict__ c,
                                       const float* __restrict__ a, float b) {
    float acc = b;
#pragma unroll
    for (int i = 0; i < NTHR; ++i) {
        acc = fmaf(a[i], stepfn(v, big[i], c[i]), acc);  // fma(clamp) + fmac
    }
    return acc;
}

__device__ __forceinline__ f4 qmap4(f4 v,
                                    const float* __restrict__ big,
                                    const float* __restrict__ c,
                                    const float* __restrict__ a, float b) {
    f4 r;
    r.x = qmap1(v.x, big, c, a, b);
    r.y = qmap1(v.y, big, c, a, b);
    r.z = qmap1(v.z, big, c, a, b);
    r.w = qmap1(v.w, big, c, a, b);
    return r;
}

__global__ __launch_bounds__(BLOCK) void quant15_fwd_kernel(
    const f4* __restrict__ x4,
    const float* __restrict__ thr,
    const float* __restrict__ alp,
    const float* __restrict__ bptr,
    f4* __restrict__ y4,
    const float* __restrict__ x_scalar,
    float* __restrict__ y_scalar,
    int n4, int rem)
{
    // Uniform quantizer parameters -> scalar loads; derive (BIG_i, c_i) once per
    // thread (amortized over 64 elements).
    float big[NTHR], c[NTHR], a[NTHR];
#pragma unroll
    for (int i = 0; i < NTHR; ++i) {
        const float t = thr[i];
        a[i] = alp[i];
        unsigned eb = (__float_as_uint(t) >> 23) & 0xFFu;  // biased exponent of t
        if (eb < 23u) eb = 23u;      // guard denormal/tiny thresholds
        // BIG = 2^(23 - e) = 2^(150 - eb) -> biased exponent field = 277 - eb.
        big[i] = __uint_as_float((277u - eb) << 23);
        c[i]   = fmaf(-t, big[i], 1.0f);   // exact integer in [-2^24+1, 2^24]
    }
    const float b = bptr[0];

    const int tid    = blockIdx.x * blockDim.x + threadIdx.x;
    const int stride = gridDim.x * blockDim.x;

    int i = tid;
    // Main unrolled-by-4 grid-stride loop: 4 B128 NT loads in flight per thread.
    for (; i + 3 * stride < n4; i += 4 * stride) {
        f4 v0 = __builtin_nontemporal_load(&x4[i]);
        f4 v1 = __builtin_nontemporal_load(&x4[i + stride]);
        f4 v2 = __builtin_nontemporal_load(&x4[i + 2 * stride]);
        f4 v3 = __builtin_nontemporal_load(&x4[i + 3 * stride]);

        // Prefetch next tile of the stream (global_prefetch_b8).
        if (i + 7 * stride < n4) {
            __builtin_prefetch((const void*)&x4[i + 4 * stride], 0, 0);
        }

        f4 r0 = qmap4(v0, big, c, a, b);
        f4 r1 = qmap4(v1, big, c, a, b);
        f4 r2 = qmap4(v2, big, c, a, b);
        f4 r3 = qmap4(v3, big, c, a, b);

        __builtin_nontemporal_store(r0, &y4[i]);
        __builtin_nontemporal_store(r1, &y4[i + stride]);
        __builtin_nontemporal_store(r2, &y4[i + 2 * stride]);
        __builtin_nontemporal_store(r3, &y4[i + 3 * stride]);
    }
    // Remainder float4s.
    for (; i < n4; i += stride) {
        f4 v = __builtin_nontemporal_load(&x4[i]);
        __builtin_nontemporal_store(qmap4(v, big, c, a, b), &y4[i]);
    }
    // Scalar tail (n % 4 elements).
    if (tid < rem) {
        const int base = n4 * 4;
        y_scalar[base + tid] = qmap1(x_scalar[base + tid], big, c, a, b);
    }
}

extern "C" void kernel_launch(void* const* d_in, const int* in_sizes, int n_in,
                              void* d_out, int out_size, void* d_ws, size_t ws_size,
                              hipStream_t stream) {
    // setup_inputs order: x, T, thresholds, alphas, b
    const float* x   = (const float*)d_in[0];
    // d_in[1] = T : only affects backward; unused in forward.
    const float* thr = (const float*)d_in[2];
    const float* alp = (const float*)d_in[3];
    const float* bp  = (const float*)d_in[4];
    float* y = (float*)d_out;

    const int n   = in_sizes[0];     // 32*4096*512 = 67,108,864
    const int n4  = n >> 2;
    const int rem = n & 3;

    int blocks = (n4 + BLOCK - 1) / BLOCK;
    if (blocks > MAX_BLOCKS) blocks = MAX_BLOCKS;
    if (blocks < 1) blocks = 1;

    quant15_fwd_kernel<<<blocks, BLOCK, 0, stream>>>(
        (const f4*)x, thr, alp, bp, (f4*)y, x, y, n4, rem);
}